// LGConv_66400194396296
// MI455X (gfx1250) — compile-verified
//
#include <hip/hip_runtime.h>
#include <hip/hip_bf16.h>
#include <stdint.h>

// ---------------------------------------------------------------------------
// LGConv scatter-add, CSR-converted for MI455X (gfx1250).
//   Phase 1: count edges per destination          (int atomics)
//   Phase 2: exclusive scan -> row offsets        (single-block LDS scan)
//   Phase 3: bucket fill (src_idx, weight) pairs  (async global->LDS staging)
//   Phase 4: per-row gather + register FMA accum  (no float atomics at all)
// ---------------------------------------------------------------------------

#define D_FEAT 64            // floats per node row
#define VEC    16            // float4 lanes per row (16 * 4 = 64)

__device__ __forceinline__ void wait_async0() {
#if __has_builtin(__builtin_amdgcn_s_wait_asynccnt)
  __builtin_amdgcn_s_wait_asynccnt(0);
#else
  asm volatile("s_wait_asynccnt 0" ::: "memory");
#endif
}

// ---------------- Phase 0: zero the per-node counters ----------------------
__global__ void lg_zero_i32(int* __restrict__ p, int n) {
  int i = blockIdx.x * 256 + threadIdx.x;
  if (i < n) p[i] = 0;
}

// ---------------- Phase 1: histogram of destination indices ----------------
__global__ void lg_count(const int* __restrict__ dsti, int* __restrict__ counts, int E) {
  int e = blockIdx.x * 256 + threadIdx.x;
  if (e < E) atomicAdd(&counts[dsti[e]], 1);
}

// ---------------- Phase 2: exclusive scan over N counts --------------------
// One block of 1024 threads; each thread serially owns ceil(N/1024) elements.
// counts[] is rewritten in place with the running cursor (= row start).
__global__ __launch_bounds__(1024)
void lg_scan(int* __restrict__ counts, int* __restrict__ offsets, int N) {
  __shared__ int sh[1024];
  int t = threadIdx.x;
  int chunk = (N + 1023) >> 10;
  int start = t * chunk;
  int end   = min(start + chunk, N);

  int sum = 0;
  for (int i = start; i < end; ++i) sum += counts[i];
  sh[t] = sum;
  __syncthreads();
  for (int off = 1; off < 1024; off <<= 1) {
    int v = (t >= off) ? sh[t - off] : 0;
    __syncthreads();
    sh[t] += v;
    __syncthreads();
  }
  int run = sh[t] - sum;                 // exclusive prefix of this chunk
  for (int i = start; i < end; ++i) {
    int c = counts[i];
    offsets[i] = run;
    counts[i]  = run;                    // becomes the fill cursor
    run += c;
  }
  if (t == 1023) offsets[N] = sh[1023];  // total edge count
}

// ---------------- Phase 3: bucket fill with async LDS staging --------------
// Each block stages a 256-edge tile of (src, dst, w) into LDS using the
// CDNA5 async global->LDS data path, waits on ASYNCcnt, then scatters the
// interleaved (src, weight) pair to its CSR slot.
__global__ __launch_bounds__(256)
void lg_fill(const int*   __restrict__ ei,      // [2*E]: src row then dst row
             const float* __restrict__ ew,      // [E]
             int*  __restrict__ cursor,         // [N] running row cursors
             int2* __restrict__ bucket,         // [E] (src_idx, w as bits)
             int E) {
  __shared__ int   sh_src[256];
  __shared__ int   sh_dst[256];
  __shared__ float sh_w[256];

  int tid = threadIdx.x;
  long e  = (long)blockIdx.x * 256 + tid;
  long ec = (e < E) ? e : (long)E - 1;          // clamp tail; issue uniformly
  unsigned voff = (unsigned)(ec * 4);           // byte offset into int32/f32 arrays

  unsigned a_src = (unsigned)(uintptr_t)&sh_src[tid];
  unsigned a_dst = (unsigned)(uintptr_t)&sh_dst[tid];
  unsigned a_w   = (unsigned)(uintptr_t)&sh_w[tid];

  const int* srcp = ei;         // edge_index row 0
  const int* dstp = ei + E;     // edge_index row 1

  // CDNA5 async copies: GVS mode (SGPR base + 32-bit VGPR byte offset),
  // tracked on ASYNCcnt, landing directly in LDS with no VGPR round-trip.
  asm volatile("global_load_async_to_lds_b32 %0, %1, %2"
               :: "v"(a_src), "v"(voff), "s"(srcp) : "memory");
  asm volatile("global_load_async_to_lds_b32 %0, %1, %2"
               :: "v"(a_dst), "v"(voff), "s"(dstp) : "memory");
  asm volatile("global_load_async_to_lds_b32 %0, %1, %2"
               :: "v"(a_w),   "v"(voff), "s"(ew)   : "memory");
  wait_async0();
  __syncthreads();

  if (e < E) {
    int d   = sh_dst[tid];
    int pos = atomicAdd(&cursor[d], 1);
    bucket[pos] = make_int2(sh_src[tid], __float_as_int(sh_w[tid]));
  }
}

// ---------------- Phase 4: gather + accumulate (no atomics) ----------------
// 16 lanes per destination node, float4 per lane (covers d=64). Software-
// pipelined: the next bucket pair is loaded and the next src row prefetched
// (global_prefetch_b8) while the current row's FMAs execute.
__global__ __launch_bounds__(256)
void lg_gather(const float4* __restrict__ srcv,     // src_x as [N][16] float4
               const int*    __restrict__ offsets,  // [N+1]
               const int2*   __restrict__ bucket,   // [E]
               float4*       __restrict__ outv,     // out as [N][16] float4
               int N) {
  int t    = threadIdx.x;
  int node = blockIdx.x * (256 / VEC) + (t >> 4);
  int lane = t & (VEC - 1);
  if (node >= N) return;

  int beg = offsets[node];
  int end = offsets[node + 1];
  float4 acc = make_float4(0.f, 0.f, 0.f, 0.f);

  if (beg < end) {
    int2 p = bucket[beg];
    for (int e = beg; e < end; ++e) {
      int2 np = (e + 1 < end) ? bucket[e + 1] : p;
      __builtin_prefetch(&srcv[(size_t)np.x * VEC + lane], 0, 0);  // global_prefetch_b8
      float  w = __int_as_float(p.y);
      float4 v = srcv[(size_t)p.x * VEC + lane];
      acc.x = fmaf(w, v.x, acc.x);
      acc.y = fmaf(w, v.y, acc.y);
      acc.z = fmaf(w, v.z, acc.z);
      acc.w = fmaf(w, v.w, acc.w);
      p = np;
    }
  }
  outv[(size_t)node * VEC + lane] = acc;   // one coalesced 256B row store
}

// ---------------- Fallback: direct atomic scatter (if ws too small) --------
__global__ void lg_zero_f32(float* __restrict__ p, long n) {
  long i = (long)blockIdx.x * 256 + threadIdx.x;
  if (i < n) p[i] = 0.f;
}

__global__ __launch_bounds__(256)
void lg_scatter_atomic(const float4* __restrict__ srcv,
                       const int*    __restrict__ ei,
                       const float*  __restrict__ ew,
                       float*        __restrict__ out, int E) {
  long idx = (long)blockIdx.x * 256 + threadIdx.x;
  if (idx >= (long)E * VEC) return;
  int e = (int)(idx >> 4);
  int g = (int)(idx & (VEC - 1));
  int   s = ei[e];
  int   d = ei[(long)E + e];
  float w = ew[e];
  float4 v = srcv[(size_t)s * VEC + g];
  float* o = out + (size_t)d * D_FEAT + g * 4;
  atomicAdd(o + 0, w * v.x);
  atomicAdd(o + 1, w * v.y);
  atomicAdd(o + 2, w * v.z);
  atomicAdd(o + 3, w * v.w);
}

// ---------------------------------------------------------------------------
extern "C" void kernel_launch(void* const* d_in, const int* in_sizes, int n_in,
                              void* d_out, int out_size, void* d_ws, size_t ws_size,
                              hipStream_t stream) {
  const float* src_x = (const float*)d_in[0];
  // d_in[1] = dst_x: only defines the output row count
  const int*   ei    = (const int*)d_in[2];    // [2, E] int32
  const float* ew    = (const float*)d_in[3];  // [E, 1]
  float*       out   = (float*)d_out;

  const int N = in_sizes[1] / D_FEAT;          // destination node count
  const int E = in_sizes[3];                   // edge count

  // Workspace layout: offsets[N+1] | cursor[N] | bucket[E] (int2, 8B-aligned)
  size_t ofs_bytes = (((size_t)(2 * N + 1) * 4) + 7) & ~(size_t)7;
  size_t need      = ofs_bytes + (size_t)E * 8;

  if (ws_size >= need) {
    int*  offsets = (int*)d_ws;
    int*  cursor  = offsets + (N + 1);
    int2* bucket  = (int2*)((char*)d_ws + ofs_bytes);

    lg_zero_i32<<<(N + 255) / 256, 256, 0, stream>>>(cursor, N);
    lg_count  <<<(E + 255) / 256, 256, 0, stream>>>(ei + E, cursor, E);
    lg_scan   <<<1, 1024, 0, stream>>>(cursor, offsets, N);
    lg_fill   <<<(E + 255) / 256, 256, 0, stream>>>(ei, ew, cursor, bucket, E);
    lg_gather <<<(N + (256 / VEC) - 1) / (256 / VEC), 256, 0, stream>>>(
        (const float4*)src_x, offsets, bucket, (float4*)out, N);
  } else {
    long nout = (long)out_size;
    lg_zero_f32<<<(int)((nout + 255) / 256), 256, 0, stream>>>(out, nout);
    long total = (long)E * VEC;
    lg_scatter_atomic<<<(int)((total + 255) / 256), 256, 0, stream>>>(
        (const float4*)src_x, ei, ew, out, E);
  }
}